// GNN_14224931684915
// MI455X (gfx1250) — compile-verified
//
#include <hip/hip_runtime.h>

#define N_NODES   100000
#define N_EDGES   1600000
#define D_IN      128
#define D_OUT     64
#define NUM_LAYERS 3
#define BN_EPS    1e-5f
#define ROWS_PER_BLK 64

typedef float v2f __attribute__((ext_vector_type(2)));
typedef float v4f __attribute__((ext_vector_type(4)));
typedef float v8f __attribute__((ext_vector_type(8)));

// ---------------------------------------------------------------------------
// Zero the aggregation buffer (N*D floats) and the 3*128 stats floats.
// ---------------------------------------------------------------------------
__global__ void k_zero(float* __restrict__ agg, float* __restrict__ stats) {
    size_t i = (size_t)blockIdx.x * blockDim.x + threadIdx.x;
    const size_t total4 = (size_t)N_NODES * D_IN / 4;   // 3.2M float4
    if (i < total4) {
        v4f z = {0.f, 0.f, 0.f, 0.f};
        ((v4f*)agg)[i] = z;
    }
    if (blockIdx.x == 0 && threadIdx.x < 3 * D_IN) stats[threadIdx.x] = 0.f;
}

// ---------------------------------------------------------------------------
// readout[f] = sum_i in[i][f]  (per-block partial sums in registers + atomic)
// blockDim.x == 128 (one thread per feature), ROWS_PER_BLK rows per block.
// ---------------------------------------------------------------------------
__global__ void k_colsum(const float* __restrict__ in, float* __restrict__ readout) {
    const int f    = threadIdx.x;
    const int row0 = blockIdx.x * ROWS_PER_BLK;
    const int rmax = min(ROWS_PER_BLK, N_NODES - row0);
    float s = 0.f;
    for (int r = 0; r < rmax; ++r)
        s += in[(size_t)(row0 + r) * D_IN + f];
    unsafeAtomicAdd(&readout[f], s);   // global_atomic_add_f32
}

// ---------------------------------------------------------------------------
// Edge scatter-add: one wave32 per edge; lane L moves features [4L, 4L+4).
// agg[dst[e]][:] += in[src[e]][:]   (L2-resident fp32 atomics)
// ---------------------------------------------------------------------------
__global__ void k_scatter(const float* __restrict__ in,
                          const long long* __restrict__ ei,
                          float* __restrict__ agg) {
    const int wave = (int)(((size_t)blockIdx.x * blockDim.x + threadIdx.x) >> 5);
    const int lane = threadIdx.x & 31;
    if (wave >= N_EDGES) return;
    const long long dst = ei[wave];             // edge_index[0] = dst
    const long long src = ei[N_EDGES + wave];   // edge_index[1] = src
    const float4 v = ((const float4*)(in + (size_t)src * D_IN))[lane]; // global_load_b128
    float* op = agg + (size_t)dst * D_IN + lane * 4;
    unsafeAtomicAdd(op + 0, v.x);
    unsafeAtomicAdd(op + 1, v.y);
    unsafeAtomicAdd(op + 2, v.z);
    unsafeAtomicAdd(op + 3, v.w);
}

// ---------------------------------------------------------------------------
// mid = in + agg + readout  (written in place over agg), accumulate
// per-feature sum / sum-of-squares for BatchNorm batch statistics.
// ---------------------------------------------------------------------------
__global__ void k_combine(const float* __restrict__ in, float* __restrict__ mid,
                          const float* __restrict__ readout,
                          float* __restrict__ sum, float* __restrict__ sumsq) {
    const int f    = threadIdx.x;
    const int row0 = blockIdx.x * ROWS_PER_BLK;
    const int rmax = min(ROWS_PER_BLK, N_NODES - row0);
    const float ro = readout[f];
    float ls = 0.f, lq = 0.f;
    for (int r = 0; r < rmax; ++r) {
        const size_t idx = (size_t)(row0 + r) * D_IN + f;
        const float v = in[idx] + mid[idx] + ro;
        mid[idx] = v;
        ls += v;
        lq += v * v;
    }
    unsafeAtomicAdd(&sum[f], ls);
    unsafeAtomicAdd(&sumsq[f], lq);
}

// ---------------------------------------------------------------------------
// BatchNorm (training stats, biased var) + ReLU:  outh = relu(mid*scale+shift)
// ---------------------------------------------------------------------------
__global__ void k_bn_relu(const float* __restrict__ mid, float* __restrict__ outh,
                          const float* __restrict__ sum, const float* __restrict__ sumsq,
                          const float* __restrict__ gamma, const float* __restrict__ beta) {
    const int f    = threadIdx.x;
    const int row0 = blockIdx.x * ROWS_PER_BLK;
    const int rmax = min(ROWS_PER_BLK, N_NODES - row0);
    const float invN  = 1.0f / (float)N_NODES;
    const float mean  = sum[f] * invN;
    const float var   = sumsq[f] * invN - mean * mean;
    const float scale = gamma[f] * rsqrtf(var + BN_EPS);
    const float shift = beta[f] - mean * scale;
    for (int r = 0; r < rmax; ++r) {
        const size_t idx = (size_t)(row0 + r) * D_IN + f;
        const float v = mid[idx] * scale + shift;
        outh[idx] = v > 0.f ? v : 0.f;
    }
}

// ---------------------------------------------------------------------------
// Final head: out[100000x64] = h[100000x128] @ W[128x64] + b, via
// V_WMMA_F32_16X16X4_F32. One wave per 16x16 output tile, K in steps of 4.
// ISA layouts (wave32):
//   A 16x4 f32 : lane(0-15)=row M, VGPR0/1 = K {0,1}; lanes 16-31 = K {2,3}
//   B 4x16 f32 : lane(0-15)=col N, VGPR0/1 = K {0,1}; lanes 16-31 = K {2,3}
//   C/D 16x16  : VGPR r -> M = r + 8*(lane>=16), N = lane&15
// ---------------------------------------------------------------------------
__global__ void k_gemm_wmma(const float* __restrict__ h, const float* __restrict__ W,
                            const float* __restrict__ bias, float* __restrict__ out) {
    const int wave  = (int)(((size_t)blockIdx.x * blockDim.x + threadIdx.x) >> 5);
    const int mtile = wave >> 2;        // 6250 M-tiles
    const int ntile = wave & 3;         // 4 N-tiles
    const int lane  = threadIdx.x & 31;
    const int lmod  = lane & 15;
    const int khalf = lane >> 4;        // 0 or 1 -> K offset 0 or 2

    const float* arow = h + (size_t)(mtile * 16 + lmod) * D_IN + khalf * 2;
    const float* bcol = W + (size_t)(khalf * 2) * D_OUT + ntile * 16 + lmod;

    v8f acc = {};
    #pragma unroll
    for (int k = 0; k < D_IN; k += 4) {
        v2f a = *(const v2f*)(arow + k);           // 8B aligned: k even, khalf*2 even
        v2f b;
        b.x = bcol[(size_t)k * D_OUT];
        b.y = bcol[(size_t)(k + 1) * D_OUT];
        acc = __builtin_amdgcn_wmma_f32_16x16x4_f32(
            /*neg_a=*/false, a, /*neg_b=*/false, b,
            /*c_mod=*/(short)0, acc, /*reuse_a=*/false, /*reuse_b=*/false);
    }

    const int col = ntile * 16 + lmod;
    const float bv = bias[col];
    float* orow = out + (size_t)(mtile * 16 + khalf * 8) * D_OUT + col;
    #pragma unroll
    for (int r = 0; r < 8; ++r)
        orow[(size_t)r * D_OUT] = acc[r] + bv;
}

// ---------------------------------------------------------------------------
extern "C" void kernel_launch(void* const* d_in, const int* in_sizes, int n_in,
                              void* d_out, int out_size, void* d_ws, size_t ws_size,
                              hipStream_t stream) {
    (void)in_sizes; (void)n_in; (void)out_size; (void)ws_size;
    const float*     x     = (const float*)d_in[0];
    const long long* ei    = (const long long*)d_in[1];   // int64 per reference
    const float*     gamma = (const float*)d_in[2];       // [3,128]
    const float*     beta  = (const float*)d_in[3];       // [3,128]
    const float*     W     = (const float*)d_in[4];       // [128,64]
    const float*     b     = (const float*)d_in[5];       // [64]
    float*           out   = (float*)d_out;               // [100000,64]

    float* hA    = (float*)d_ws;                          // normalized h   (N*D)
    float* hB    = hA + (size_t)N_NODES * D_IN;           // agg / mid      (N*D)
    float* stats = hB + (size_t)N_NODES * D_IN;           // 3*128 floats
    float* readout = stats;
    float* sum     = stats + D_IN;
    float* sumsq   = stats + 2 * D_IN;

    const int ZB = (N_NODES * D_IN / 4 + 255) / 256;      // 12500 blocks
    const int RB = (N_NODES + ROWS_PER_BLK - 1) / ROWS_PER_BLK; // 1563 blocks
    const int SB = N_EDGES / 8;                           // 200000 blocks (8 waves/blk)

    for (int l = 0; l < NUM_LAYERS; ++l) {
        const float* in = (l == 0) ? x : hA;
        k_zero   <<<ZB, 256, 0, stream>>>(hB, stats);
        k_colsum <<<RB, 128, 0, stream>>>(in, readout);
        k_scatter<<<SB, 256, 0, stream>>>(in, ei, hB);
        k_combine<<<RB, 128, 0, stream>>>(in, hB, readout, sum, sumsq);
        k_bn_relu<<<RB, 128, 0, stream>>>(hB, hA, sum, sumsq,
                                          gamma + l * D_IN, beta + l * D_IN);
    }
    // 6250*4 = 25000 tiles, 8 waves/block -> 3125 blocks, no remainder
    k_gemm_wmma<<<(6250 * 4) / 8, 256, 0, stream>>>(hA, W, b, out);
}